// KG_diverse_103079215780
// MI455X (gfx1250) — compile-verified
//
#include <hip/hip_runtime.h>
#include <math.h>

#define NU 50000
#define NI 30000
#define NE 80000
#define DIM 64
#define NKG 2000000
#define NINT 1000000
#define NADJ 2000000
#define EPS_PD 1e-6f
#define INV_TEMP 5.0f

typedef __attribute__((ext_vector_type(2))) float v2f;
typedef __attribute__((ext_vector_type(8))) float v8f;

// ---------------- utility kernels ----------------

__global__ void k_zero(float* __restrict__ p, long long n) {
  long long i = blockIdx.x * (long long)blockDim.x + threadIdx.x;
  long long s = (long long)gridDim.x * blockDim.x;
  for (; i < n; i += s) p[i] = 0.0f;
}

__global__ void k_add(float* __restrict__ a, const float* __restrict__ b, long long n) {
  long long i = blockIdx.x * (long long)blockDim.x + threadIdx.x;
  if (i < n) a[i] += b[i];
}

__global__ void k_scale(float* __restrict__ a, float s, long long n) {
  long long i = blockIdx.x * (long long)blockDim.x + threadIdx.x;
  if (i < n) a[i] *= s;
}

// ---------------- KG aggregation ----------------

__global__ void k_count(const int* __restrict__ head, float* __restrict__ cnt, int n) {
  int i = blockIdx.x * blockDim.x + threadIdx.x;
  if (i < n) atomicAdd(&cnt[head[i]], 1.0f);
}

// one wave per edge, lane handles a float2 slice of the 64-wide row
__global__ void k_kg_scatter(const int* __restrict__ head, const int* __restrict__ tail,
                             const float* __restrict__ e, float* __restrict__ agg, int n) {
  long long t = blockIdx.x * (long long)blockDim.x + threadIdx.x;
  int edge = (int)(t >> 5);
  if (edge >= n) return;
  int lane = (int)(t & 31);
  int h = head[edge], tl = tail[edge];
  float2 v = ((const float2*)(e + (long long)tl * DIM))[lane];
  float* dst = agg + (long long)h * DIM + lane * 2;
  atomicAdd(dst + 0, v.x);
  atomicAdd(dst + 1, v.y);
}

__global__ void k_div_cnt(float* __restrict__ agg, const float* __restrict__ cnt, long long n) {
  long long i = blockIdx.x * (long long)blockDim.x + threadIdx.x;
  if (i < n) agg[i] /= fmaxf(cnt[i >> 6], 1.0f);
}

// y[row] += val * x[col], one wave per nnz
__global__ void k_spmm_scatter(const int* __restrict__ rows, const int* __restrict__ cols,
                               const float* __restrict__ vals, const float* __restrict__ x,
                               float* __restrict__ y, int nnz) {
  long long t = blockIdx.x * (long long)blockDim.x + threadIdx.x;
  int i = (int)(t >> 5);
  if (i >= nnz) return;
  int lane = (int)(t & 31);
  int r = rows[i], c = cols[i];
  float v = vals[i];
  float2 xv = ((const float2*)(x + (long long)c * DIM))[lane];
  float* dst = y + (long long)r * DIM + lane * 2;
  atomicAdd(dst + 0, v * xv.x);
  atomicAdd(dst + 1, v * xv.y);
}

// ---------------- attention ----------------

__global__ void k_score(const int* __restrict__ rows, const int* __restrict__ cols,
                        const float* __restrict__ item_agg, const float* __restrict__ user_mean,
                        float* __restrict__ score, int nnz) {
  long long t = blockIdx.x * (long long)blockDim.x + threadIdx.x;
  int i = (int)(t >> 5);
  if (i >= nnz) return;
  int lane = (int)(t & 31);
  int r = rows[i], c = cols[i];
  float2 a = ((const float2*)(item_agg + (long long)c * DIM))[lane];
  float2 u = ((const float2*)(user_mean + (long long)r * DIM))[lane];
  float dx = a.x - u.x + EPS_PD;
  float dy = a.y - u.y + EPS_PD;
  float n2 = dx * dx + dy * dy;
  #pragma unroll
  for (int o = 16; o; o >>= 1) n2 += __shfl_xor(n2, o, 32);
  if (lane == 0) score[i] = sqrtf(n2) * INV_TEMP;
}

// scores are >= 0, so int-bit atomicMax == float max; mbuf pre-zeroed
__global__ void k_segmax(const float* __restrict__ score, const int* __restrict__ rows,
                         float* __restrict__ mbuf, int nnz) {
  int i = blockIdx.x * blockDim.x + threadIdx.x;
  if (i < nnz) atomicMax((int*)(mbuf + rows[i]), __float_as_int(score[i]));
}

__global__ void k_expsum(float* __restrict__ score, const int* __restrict__ rows,
                         const float* __restrict__ mbuf, float* __restrict__ denom, int nnz) {
  int i = blockIdx.x * blockDim.x + threadIdx.x;
  if (i >= nnz) return;
  int r = rows[i];
  float ex = expf(score[i] - mbuf[r]);
  score[i] = ex;  // keep numerator for the scatter pass
  atomicAdd(&denom[r], ex);
}

__global__ void k_uagg(const int* __restrict__ rows, const int* __restrict__ cols,
                       const float* __restrict__ ex, const float* __restrict__ denom,
                       const float* __restrict__ item_agg, float* __restrict__ uagg, int nnz) {
  long long t = blockIdx.x * (long long)blockDim.x + threadIdx.x;
  int i = (int)(t >> 5);
  if (i >= nnz) return;
  int lane = (int)(t & 31);
  int r = rows[i], c = cols[i];
  float att = ex[i] / denom[r];
  float2 xv = ((const float2*)(item_agg + (long long)c * DIM))[lane];
  float* dst = uagg + (long long)r * DIM + lane * 2;
  atomicAdd(dst + 0, att * xv.x);
  atomicAdd(dst + 1, att * xv.y);
}

// ---------------- WMMA row-normalization ----------------
// One wave per 16 rows. Lane L owns row L%16, columns {4c + 2*(L/16), +1}.
// With this mapping the A-matrix (16x4) and B-matrix (4x16) f32 WMMA layouts
// coincide for B = A^T, so wmma(a, a, acc) over sixteen K=4 chunks produces
// the 16x16 Gram matrix X*X^T; its diagonal is the squared row norms.
__global__ __launch_bounds__(32)
void k_norm_wmma(const float* __restrict__ x, float* __restrict__ out_norm,
                 float* __restrict__ res) {
  int lane = threadIdx.x;
  int rsel = lane & 15;
  int half = lane >> 4;
  long long row = (long long)blockIdx.x * 16 + rsel;
  const float* xr = x + row * DIM;

  v2f a[16];
  #pragma unroll
  for (int c = 0; c < 16; ++c) {
    int col = 4 * c + 2 * half;
    a[c].x = xr[col];
    a[c].y = xr[col + 1];
  }

  v8f g = {};
  #pragma unroll
  for (int c = 0; c < 16; ++c)
    g = __builtin_amdgcn_wmma_f32_16x16x4_f32(false, a[c], false, a[c],
                                              (short)0, g, false, false);

  // diagonal of C: vgpr r lane r -> C[r][r]; vgpr r lane r+24 -> C[r+8][r+8]
  float n2 = 0.0f;
  #pragma unroll
  for (int r = 0; r < 8; ++r) {
    float vlo = __shfl(g[r], r, 32);
    float vhi = __shfl(g[r], r + 24, 32);
    if (rsel == r) n2 = vlo;
    if (rsel == r + 8) n2 = vhi;
  }

  float scale = 1.0f / fmaxf(sqrtf(n2), 1e-12f);
  float* on = out_norm + row * DIM;
  float* rr = res + row * DIM;
  #pragma unroll
  for (int c = 0; c < 16; ++c) {
    int col = 4 * c + 2 * half;
    float y0 = a[c].x * scale, y1 = a[c].y * scale;
    on[col] = y0;
    on[col + 1] = y1;
    rr[col] += y0;
    rr[col + 1] += y1;
  }
}

// ---------------- host side ----------------

static inline unsigned blocks_for(long long threads, int bs) {
  return (unsigned)((threads + bs - 1) / bs);
}

extern "C" void kernel_launch(void* const* d_in, const int* in_sizes, int n_in,
                              void* d_out, int out_size, void* d_ws, size_t ws_size,
                              hipStream_t stream) {
  const float* all_embed     = (const float*)d_in[0];
  const float* interact_vals = (const float*)d_in[1];
  const float* adj_vals      = (const float*)d_in[2];
  const int*   edge_index    = (const int*)d_in[3];
  // d_in[4] = edge_type: unused by the reference forward
  const int*   interact_rows = (const int*)d_in[5];
  const int*   interact_cols = (const int*)d_in[6];
  const int*   adj_rows      = (const int*)d_in[7];
  const int*   adj_cols      = (const int*)d_in[8];
  const int* head = edge_index;
  const int* tail = edge_index + NKG;

  float* ws = (float*)d_ws;
  float* e_cur  = ws;  ws += (long long)NE * DIM;   // current entity embeddings
  float* e_nxt  = ws;  ws += (long long)NE * DIM;   // entity_agg / ping-pong
  float* e_res  = ws;  ws += (long long)NE * DIM;
  float* u_res  = ws;  ws += (long long)NU * DIM;
  float* u_mean = ws;  ws += (long long)NU * DIM;   // also scratch for user norm output
  float* u_agg  = ws;  ws += (long long)NU * DIM;
  float* score  = ws;  ws += NINT;
  float* mbuf   = ws;  ws += NU;
  float* denom  = ws;  ws += NU;
  float* cnt    = ws;  ws += NE;

  auto zero = [&](float* p, long long n) {
    k_zero<<<2048, 256, 0, stream>>>(p, n);
  };

  // residual/state init
  hipMemcpyAsync(u_res, all_embed, (size_t)NU * DIM * sizeof(float),
                 hipMemcpyDeviceToDevice, stream);
  hipMemcpyAsync(e_res, all_embed + (long long)NU * DIM, (size_t)NE * DIM * sizeof(float),
                 hipMemcpyDeviceToDevice, stream);
  hipMemcpyAsync(e_cur, all_embed + (long long)NU * DIM, (size_t)NE * DIM * sizeof(float),
                 hipMemcpyDeviceToDevice, stream);

  zero(cnt, NE);
  k_count<<<blocks_for(NKG, 256), 256, 0, stream>>>(head, cnt, NKG);

  for (int hop = 0; hop < 2; ++hop) {
    // entity_agg = segment_sum(e[tail], head) / cnt
    zero(e_nxt, (long long)NE * DIM);
    k_kg_scatter<<<blocks_for((long long)NKG * 32, 256), 256, 0, stream>>>(
        head, tail, e_cur, e_nxt, NKG);
    k_div_cnt<<<blocks_for((long long)NE * DIM, 256), 256, 0, stream>>>(
        e_nxt, cnt, (long long)NE * DIM);

    // user_mean = spmm(interactions, item_agg)  (item_agg = e_nxt[:NI])
    zero(u_mean, (long long)NU * DIM);
    k_spmm_scatter<<<blocks_for((long long)NINT * 32, 256), 256, 0, stream>>>(
        interact_rows, interact_cols, interact_vals, e_nxt, u_mean, NINT);

    // attention scores + segment softmax
    k_score<<<blocks_for((long long)NINT * 32, 256), 256, 0, stream>>>(
        interact_rows, interact_cols, e_nxt, u_mean, score, NINT);
    zero(mbuf, NU);
    k_segmax<<<blocks_for(NINT, 256), 256, 0, stream>>>(score, interact_rows, mbuf, NINT);
    zero(denom, NU);
    k_expsum<<<blocks_for(NINT, 256), 256, 0, stream>>>(score, interact_rows, mbuf, denom, NINT);

    // user_agg = segment_sum(att * item_agg[cols])
    zero(u_agg, (long long)NU * DIM);
    k_uagg<<<blocks_for((long long)NINT * 32, 256), 256, 0, stream>>>(
        interact_rows, interact_cols, score, denom, e_nxt, u_agg, NINT);

    // normalize + residual accumulation (WMMA Gram diagonal)
    k_norm_wmma<<<NE / 16, 32, 0, stream>>>(e_nxt, e_cur, e_res);
    k_norm_wmma<<<NU / 16, 32, 0, stream>>>(u_agg, u_mean, u_res);
  }

  // LightGCN: x = concat(u_res, e_res[:NI]); acc in d_out; ping-pong in e_cur/e_nxt
  float* xa = e_cur;
  float* xb = e_nxt;
  hipMemcpyAsync(xa, u_res, (size_t)NU * DIM * sizeof(float),
                 hipMemcpyDeviceToDevice, stream);
  hipMemcpyAsync(xa + (long long)NU * DIM, e_res, (size_t)NI * DIM * sizeof(float),
                 hipMemcpyDeviceToDevice, stream);
  hipMemcpyAsync(d_out, xa, (size_t)(NU + NI) * DIM * sizeof(float),
                 hipMemcpyDeviceToDevice, stream);

  const long long NX = (long long)(NU + NI) * DIM;
  for (int l = 0; l < 2; ++l) {
    zero(xb, NX);
    k_spmm_scatter<<<blocks_for((long long)NADJ * 32, 256), 256, 0, stream>>>(
        adj_rows, adj_cols, adj_vals, xa, xb, NADJ);
    k_add<<<blocks_for(NX, 256), 256, 0, stream>>>((float*)d_out, xb, NX);
    float* t = xa; xa = xb; xb = t;
  }
  k_scale<<<blocks_for(NX, 256), 256, 0, stream>>>((float*)d_out, 1.0f / 3.0f, NX);
}